// GNNModelSimple_66030827208836
// MI455X (gfx1250) — compile-verified
//
#include <hip/hip_runtime.h>
#include <hip/hip_bf16.h>
#include <math.h>

typedef __attribute__((ext_vector_type(2))) float v2f;
typedef __attribute__((ext_vector_type(8))) float v8f;

#define N_NODES  200000
#define N_EDGES  6400000
#define N_GRAPHS 4096
#define IN_CH    128
#define HID      12
#define HPAD     16
#define NTILES   (N_NODES / 16)   // 12500, exact

// ---------------------------------------------------------------- utilities
__global__ void zero_f(float* __restrict__ p, int n) {
  int i = blockIdx.x * blockDim.x + threadIdx.x;
  int stride = gridDim.x * blockDim.x;
  for (; i < n; i += stride) p[i] = 0.0f;
}

__global__ void deg_count(const int* __restrict__ col, float* __restrict__ deg, int E) {
  int e = blockIdx.x * blockDim.x + threadIdx.x;
  if (e < E) atomicAdd(&deg[col[e]], 1.0f);
}

__global__ void make_dinv(float* __restrict__ deg, int n) {
  int i = blockIdx.x * blockDim.x + threadIdx.x;
  if (i < n) deg[i] = 1.0f / sqrtf(deg[i] + 1.0f);   // +1 self loop, D^-1/2
}

// ---------------------------------------------------------------- GEMM1: [N,128] @ [128,12] -> h[N,16]
// One wave per 16-node tile. A tile staged in LDS (coalesced float4). W1 staged once per
// block, transposed + zero-padded, so the inner loop is ds_load_b64 + ds_load_b64 + v_wmma
// with no predication (EXEC stays all-ones around every WMMA).
__global__ void __launch_bounds__(128)
gemm1_wmma(const float* __restrict__ x, const float* __restrict__ W1, float* __restrict__ h) {
  __shared__ float4 ldsbuf[4 * 512];                  // 4 waves * 16 rows * 128 cols (32 KB)
  __shared__ __align__(16) float w1t[HPAD * IN_CH];   // w1t[n][k], rows n>=12 are zero (8 KB)
  const int tid  = threadIdx.x;
  const int lane = tid & 31;
  const int wave = tid >> 5;
  const int tile = blockIdx.x * 4 + wave;
  const bool active = tile < NTILES;                  // wave-uniform

  // stage W1 transposed + padded (once per block)
#pragma unroll
  for (int i = 0; i < (HPAD * IN_CH) / 128; ++i) {
    const int flat = tid + 128 * i;
    const int k = flat >> 4;                          // 0..127
    const int n = flat & 15;                          // 0..15
    w1t[n * IN_CH + k] = (n < HID) ? W1[k * HID + n] : 0.0f;
  }
  // stage this wave's 16x128 A tile
  if (active) {
    const float4* src = (const float4*)(x + (long long)tile * 16 * IN_CH);
    float4* dst = ldsbuf + wave * 512;
#pragma unroll
    for (int i = 0; i < 16; ++i)
      dst[lane + 32 * i] = src[lane + 32 * i];        // coalesced
  }
  __syncthreads();
  if (!active) return;

  const int base = tile * 16;
  const int g  = lane >> 4;                           // half-wave group
  const int mr = lane & 15;                           // A row / B-D column
  const float* arow = (const float*)(ldsbuf + wave * 512) + mr * IN_CH;
  const float* brow = w1t + mr * IN_CH;

  v8f acc = {};
#pragma unroll 8
  for (int k0 = 0; k0 < IN_CH; k0 += 4) {
    const int kk = k0 + 2 * g;
    v2f a = *(const v2f*)(arow + kk);                 // A[mr][kk..kk+1]
    v2f b = *(const v2f*)(brow + kk);                 // B[kk..kk+1][mr] (transposed store)
    acc = __builtin_amdgcn_wmma_f32_16x16x4_f32(false, a, false, b,
                                                (short)0, acc, false, false);
  }
#pragma unroll
  for (int v = 0; v < 8; ++v)                         // D: VGPR v -> row v (+8 upper half)
    h[(long long)(base + v + 8 * g) * HPAD + mr] = acc[v];
}

// ---------------------------------------------------------------- GEMM2: [N,16pad] @ [12,12] -> h[N,16]
__global__ void __launch_bounds__(128)
gemm2_wmma(const float* __restrict__ hin, const float* __restrict__ W2, float* __restrict__ hout) {
  __shared__ __align__(16) float w2t[HPAD * HPAD];    // w2t[n][k], zero-padded (1 KB)
  const int tid  = threadIdx.x;
  const int lane = tid & 31;
  const int wave = tid >> 5;
  const int tile = blockIdx.x * 4 + wave;
  const bool active = tile < NTILES;                  // wave-uniform

#pragma unroll
  for (int i = 0; i < (HPAD * HPAD) / 128; ++i) {
    const int flat = tid + 128 * i;
    const int k = flat >> 4;
    const int n = flat & 15;
    w2t[n * HPAD + k] = (n < HID && k < HID) ? W2[k * HID + n] : 0.0f;
  }
  __syncthreads();
  if (!active) return;

  const int base = tile * 16;
  const int g  = lane >> 4;
  const int mr = lane & 15;
  const float* arow = hin + (long long)(base + mr) * HPAD;
  const float* brow = w2t + mr * HPAD;

  v8f acc = {};
#pragma unroll
  for (int k0 = 0; k0 < HPAD; k0 += 4) {
    const int kk = k0 + 2 * g;
    v2f a = *(const v2f*)(arow + kk);
    v2f b = *(const v2f*)(brow + kk);
    acc = __builtin_amdgcn_wmma_f32_16x16x4_f32(false, a, false, b,
                                                (short)0, acc, false, false);
  }
#pragma unroll
  for (int v = 0; v < 8; ++v)
    hout[(long long)(base + v + 8 * g) * HPAD + mr] = acc[v];
}

// ---------------------------------------------------------------- edge gather*norm -> atomic scatter
// 4 lanes per edge (q=0..2 carry 4 channels each; q==3 idle keeps float4 alignment).
__global__ void edge_agg(const int* __restrict__ row, const int* __restrict__ col,
                         const float* __restrict__ dinv, const float* __restrict__ h,
                         float* __restrict__ agg) {
  unsigned int t = blockIdx.x * blockDim.x + threadIdx.x;
  unsigned int e = t >> 2;
  unsigned int q = t & 3;
  if (e >= N_EDGES || q >= 3) return;
  const int r = row[e], c = col[e];
  const float w = dinv[r] * dinv[c];
  const float4 hv = *(const float4*)(h + (long long)r * HPAD + q * 4);
  float* dst = agg + (long long)c * HPAD + q * 4;
  atomicAdd(dst + 0, hv.x * w);
  atomicAdd(dst + 1, hv.y * w);
  atomicAdd(dst + 2, hv.z * w);
  atomicAdd(dst + 3, hv.w * w);
}

// ---------------------------------------------------------------- self-loop + bias + leaky_relu (in place)
__global__ void finish_act(float* __restrict__ agg, const float* __restrict__ h,
                           const float* __restrict__ dinv, const float* __restrict__ bias) {
  unsigned int t = blockIdx.x * blockDim.x + threadIdx.x;
  unsigned int node = t >> 2;
  unsigned int q = t & 3;
  if (node >= N_NODES) return;
  float d2 = dinv[node]; d2 *= d2;
  float4 av = *(float4*)(agg + (long long)node * HPAD + q * 4);
  const float4 hv = *(const float4*)(h + (long long)node * HPAD + q * 4);
  float vv[4] = {av.x + hv.x * d2, av.y + hv.y * d2, av.z + hv.z * d2, av.w + hv.w * d2};
#pragma unroll
  for (int j = 0; j < 4; ++j) {
    const int c = q * 4 + j;
    float v = vv[j] + (c < HID ? bias[c] : 0.0f);   // pad cols stay exactly 0
    vv[j] = v > 0.0f ? v : 0.01f * v;
  }
  av.x = vv[0]; av.y = vv[1]; av.z = vv[2]; av.w = vv[3];
  *(float4*)(agg + (long long)node * HPAD + q * 4) = av;
}

// ---------------------------------------------------------------- global_add_pool
__global__ void pool(const float* __restrict__ h, const int* __restrict__ batch,
                     float* __restrict__ xg) {
  unsigned int t = blockIdx.x * blockDim.x + threadIdx.x;
  unsigned int node = t >> 2;
  unsigned int q = t & 3;
  if (node >= N_NODES || q >= 3) return;
  const int gph = batch[node];
  const float4 hv = *(const float4*)(h + (long long)node * HPAD + q * 4);
  float* dst = xg + (long long)gph * HPAD + q * 4;
  atomicAdd(dst + 0, hv.x);
  atomicAdd(dst + 1, hv.y);
  atomicAdd(dst + 2, hv.z);
  atomicAdd(dst + 3, hv.w);
}

// ---------------------------------------------------------------- attention head + output linear
__global__ void head(const float* __restrict__ xg,
                     const float* __restrict__ Wa, const float* __restrict__ ba,
                     const float* __restrict__ Wb, const float* __restrict__ bb,
                     const float* __restrict__ Wc, const float* __restrict__ bc,
                     const float* __restrict__ Wout, const float* __restrict__ bout,
                     float* __restrict__ out) {
  int g = blockIdx.x * blockDim.x + threadIdx.x;
  if (g >= N_GRAPHS) return;
  float xv[HID];
#pragma unroll
  for (int k = 0; k < HID; ++k) xv[k] = xg[(long long)g * HPAD + k];

  float A[4];
#pragma unroll
  for (int j = 0; j < 4; ++j) {
    float sa = ba[j], sb = bb[j];
#pragma unroll
    for (int k = 0; k < HID; ++k) {
      sa += xv[k] * Wa[k * 4 + j];
      sb += xv[k] * Wb[k * 4 + j];
    }
    A[j] = tanhf(sa) * (1.0f / (1.0f + expf(-sb)));
  }
  float z = bc[0];
#pragma unroll
  for (int j = 0; j < 4; ++j) z += A[j] * Wc[j];
  const float att = expf(z - z);                 // softmax over size-1 axis == 1.0

#pragma unroll
  for (int o = 0; o < 2; ++o) {
    float s = bout[o] + att * Wout[o];           // feat[0] = att
#pragma unroll
    for (int k = 0; k < HID; ++k) s += xv[k] * Wout[(1 + k) * 2 + o];
    out[g * 2 + o] = s;
  }
}

// ---------------------------------------------------------------- launch
extern "C" void kernel_launch(void* const* d_in, const int* in_sizes, int n_in,
                              void* d_out, int out_size, void* d_ws, size_t ws_size,
                              hipStream_t stream) {
  (void)in_sizes; (void)n_in; (void)out_size; (void)ws_size;
  const float* x     = (const float*)d_in[0];
  const int*   ei    = (const int*)d_in[1];
  const int*   batch = (const int*)d_in[2];
  const float* W1    = (const float*)d_in[3];
  const float* b1    = (const float*)d_in[4];
  const float* W2    = (const float*)d_in[5];
  const float* b2    = (const float*)d_in[6];
  const float* Wa    = (const float*)d_in[7];
  const float* ba    = (const float*)d_in[8];
  const float* Wb    = (const float*)d_in[9];
  const float* bb    = (const float*)d_in[10];
  const float* Wc    = (const float*)d_in[11];
  const float* bc    = (const float*)d_in[12];
  const float* Wout  = (const float*)d_in[13];
  const float* bout  = (const float*)d_in[14];
  float* out = (float*)d_out;
  const int* row = ei;
  const int* col = ei + N_EDGES;

  // workspace layout (floats): deg[N] | agg[N*16] | h[N*16] | xg[G*16]  (~26.7 MB)
  float* ws  = (float*)d_ws;
  float* deg = ws;
  float* agg = ws + N_NODES;
  float* h   = agg + (size_t)N_NODES * HPAD;
  float* xg  = h + (size_t)N_NODES * HPAD;

  const int B = 256;
  const int gemm_blocks = (NTILES + 3) / 4;
  const int edge4_blocks = (int)(((long long)N_EDGES * 4 + B - 1) / B);
  const int node4_blocks = (N_NODES * 4 + B - 1) / B;

  // init accumulators (harness does not re-zero between replays)
  zero_f<<<2048, B, 0, stream>>>(deg, N_NODES);
  zero_f<<<4096, B, 0, stream>>>(agg, N_NODES * HPAD);
  zero_f<<<64,   B, 0, stream>>>(xg,  N_GRAPHS * HPAD);

  // degrees -> D^-1/2
  deg_count<<<(N_EDGES + B - 1) / B, B, 0, stream>>>(col, deg, N_EDGES);
  make_dinv<<<(N_NODES + B - 1) / B, B, 0, stream>>>(deg, N_NODES);

  // ---- layer 1
  gemm1_wmma<<<gemm_blocks, 128, 0, stream>>>(x, W1, h);
  edge_agg<<<edge4_blocks, B, 0, stream>>>(row, col, deg, h, agg);
  finish_act<<<node4_blocks, B, 0, stream>>>(agg, h, deg, b1);   // agg := leaky(h1)

  // ---- layer 2
  gemm2_wmma<<<gemm_blocks, 128, 0, stream>>>(agg, W2, h);       // h := h2
  zero_f<<<4096, B, 0, stream>>>(agg, N_NODES * HPAD);
  edge_agg<<<edge4_blocks, B, 0, stream>>>(row, col, deg, h, agg);
  finish_act<<<node4_blocks, B, 0, stream>>>(agg, h, deg, b2);   // agg := leaky(h2)

  // ---- pool + head
  pool<<<node4_blocks, B, 0, stream>>>(agg, batch, xg);
  head<<<(N_GRAPHS + 63) / 64, 64, 0, stream>>>(xg, Wa, ba, Wb, bb, Wc, bc, Wout, bout, out);
}